// GraphSAGEParcelModel_9818295239157
// MI455X (gfx1250) — compile-verified
//
#include <hip/hip_runtime.h>
#include <hip/hip_bf16.h>
#include <stdint.h>

#define N_NODES 100000
#define N_EDGES 1600000
#define D       128
#define LN_EPS   1e-5f
#define NORM_EPS 1e-12f

typedef __attribute__((ext_vector_type(2))) float v2f;
typedef __attribute__((ext_vector_type(8))) float v8f;

// ---------------------------------------------------------------------------
// Scatter: one wave per edge. 32 lanes x float4 = full 128-float feature row.
// Gathers are fully coalesced (512B per wave); scatter is f32 atomics that
// land in L2 (summed is 51MB, L2-resident on MI455X's 192MB L2).
// ---------------------------------------------------------------------------
__global__ __launch_bounds__(256) void scatter_sum_kernel(
    const float* __restrict__ h, const long long* __restrict__ src,
    const long long* __restrict__ dst, float* __restrict__ summed,
    float* __restrict__ cnt)
{
  int tid  = blockIdx.x * 256 + threadIdx.x;
  int edge = tid >> 5;
  int part = tid & 31;
  if (edge >= N_EDGES) return;
  int s = (int)src[edge];
  int d = (int)dst[edge];
  const float4* xin = (const float4*)(h + (size_t)s * D);
  float4 v = xin[part];
  float* out = summed + (size_t)d * D + part * 4;
  atomicAdd(out + 0, v.x);
  atomicAdd(out + 1, v.y);
  atomicAdd(out + 2, v.z);
  atomicAdd(out + 3, v.w);
  if (part == 0) atomicAdd(cnt + d, 1.0f);
}

// ---------------------------------------------------------------------------
// Fused per-layer dense kernel:
//   mean = summed / max(cnt,1)
//   t    = mean @ Wl + bl + h @ Wr          (fp32 WMMA 16x16x4, K=128)
//   y    = LayerNorm(t) * g + b
//   out  = FINAL ? y/max(||y||2, eps) : elu(y)
//
// Block = 256 threads = 8 waves; block owns 16 node rows, each wave owns a
// 16-wide slice of the 128 output columns (64 v_wmma per wave). A fragments
// come from LDS staging; B fragments stream from the 64KB weight matrices
// (L2 / WGP$ resident). C tile goes through LDS for the row-wide LN reduce.
//
// fp32 A fragment (16x4): lane L -> row = L%16, K = kbase + 2*(L/16) + {0,1}
// fp32 B fragment (4x16): lane L -> col = L%16, same K split (mirrors C/D).
// fp32 C tile (16x16):    vgpr r -> row r (lanes 0-15) / row r+8 (lanes 16-31)
// ---------------------------------------------------------------------------
template<int FINAL>
__global__ __launch_bounds__(256) void sage_gemm_ln_kernel(
    const float* __restrict__ summed, const float* __restrict__ cnt,
    const float* __restrict__ h, const float* __restrict__ Wl,
    const float* __restrict__ bl, const float* __restrict__ Wr,
    const float* __restrict__ g, const float* __restrict__ b,
    float* __restrict__ out)
{
  __shared__ float sMean[16][D];
  __shared__ float sX[16][D];
  __shared__ float sOut[16][D + 4];

  const int t    = threadIdx.x;
  const int lane = t & 31;
  const int wave = t >> 5;              // 0..7 -> N tile
  const int row0 = blockIdx.x * 16;

  // Stage A matrices (coalesced): 2048 floats each, 8 per thread.
  for (int i = 0; i < 8; ++i) {
    int idx  = t + i * 256;
    int r    = idx >> 7;
    int k    = idx & 127;
    int node = row0 + r;
    float inv = 1.0f / fmaxf(cnt[node], 1.0f);
    sMean[r][k] = summed[(size_t)node * D + k] * inv;
    sX[r][k]    = h[(size_t)node * D + k];
  }
  __syncthreads();

  const int arow  = lane & 15;
  const int khalf = (lane >> 4) << 1;           // 0 or 2
  const int ncol  = (wave << 4) + (lane & 15);  // global output column

  v8f c = {};
  for (int kk = 0; kk < D; kk += 4) {
    v2f am, ax, bL, bR;
    am.x = sMean[arow][kk + khalf];
    am.y = sMean[arow][kk + khalf + 1];
    ax.x = sX[arow][kk + khalf];
    ax.y = sX[arow][kk + khalf + 1];
    bL.x = Wl[(size_t)(kk + khalf)     * D + ncol];
    bL.y = Wl[(size_t)(kk + khalf + 1) * D + ncol];
    bR.x = Wr[(size_t)(kk + khalf)     * D + ncol];
    bR.y = Wr[(size_t)(kk + khalf + 1) * D + ncol];
    c = __builtin_amdgcn_wmma_f32_16x16x4_f32(false, am, false, bL,
                                              (short)0, c, false, false);
    c = __builtin_amdgcn_wmma_f32_16x16x4_f32(false, ax, false, bR,
                                              (short)0, c, false, false);
  }

  // Spill C tile (+ bias) to LDS for the row-wide LayerNorm.
  float bias  = bl[ncol];
  int   rbase = (lane < 16) ? 0 : 8;
#pragma unroll
  for (int r = 0; r < 8; ++r)
    sOut[rbase + r][ncol] = c[r] + bias;
  __syncthreads();

  // LayerNorm: 16 threads per row (aligned 16-lane groups -> shfl_xor safe).
  int row = t >> 4;
  int sub = t & 15;
  float v[8];
  float sum = 0.f;
#pragma unroll
  for (int j = 0; j < 8; ++j) { v[j] = sOut[row][sub * 8 + j]; sum += v[j]; }
#pragma unroll
  for (int m = 8; m >= 1; m >>= 1) sum += __shfl_xor(sum, m, 32);
  float mu  = sum * (1.0f / D);
  float var = 0.f;
#pragma unroll
  for (int j = 0; j < 8; ++j) { float dv = v[j] - mu; var += dv * dv; }
#pragma unroll
  for (int m = 8; m >= 1; m >>= 1) var += __shfl_xor(var, m, 32);
  float rstd = rsqrtf(var * (1.0f / D) + LN_EPS);

  float y[8];
#pragma unroll
  for (int j = 0; j < 8; ++j) {
    int col = sub * 8 + j;
    y[j] = (v[j] - mu) * rstd * g[col] + b[col];
  }

  float* orow = out + (size_t)(row0 + row) * D;
  if (FINAL) {
    float sq = 0.f;
#pragma unroll
    for (int j = 0; j < 8; ++j) sq += y[j] * y[j];
#pragma unroll
    for (int m = 8; m >= 1; m >>= 1) sq += __shfl_xor(sq, m, 32);
    float inv = 1.0f / fmaxf(sqrtf(sq), NORM_EPS);
#pragma unroll
    for (int j = 0; j < 8; ++j) orow[sub * 8 + j] = y[j] * inv;
  } else {
#pragma unroll
    for (int j = 0; j < 8; ++j) {
      float e = (y[j] > 0.f) ? y[j] : (__expf(y[j]) - 1.0f);
      orow[sub * 8 + j] = e;
    }
  }
}

// ---------------------------------------------------------------------------
extern "C" void kernel_launch(void* const* d_in, const int* in_sizes, int n_in,
                              void* d_out, int out_size, void* d_ws, size_t ws_size,
                              hipStream_t stream)
{
  const float*     x   = (const float*)d_in[0];
  const long long* ei  = (const long long*)d_in[1];   // int64 [2, E]
  const long long* src = ei;
  const long long* dst = ei + N_EDGES;

  const float* Wl[3] = {(const float*)d_in[2],  (const float*)d_in[7],  (const float*)d_in[12]};
  const float* bl[3] = {(const float*)d_in[3],  (const float*)d_in[8],  (const float*)d_in[13]};
  const float* Wr[3] = {(const float*)d_in[4],  (const float*)d_in[9],  (const float*)d_in[14]};
  const float* gg[3] = {(const float*)d_in[5],  (const float*)d_in[10], (const float*)d_in[15]};
  const float* bb[3] = {(const float*)d_in[6],  (const float*)d_in[11], (const float*)d_in[16]};

  char*  ws        = (char*)d_ws;
  size_t featBytes = (size_t)N_NODES * D * sizeof(float);        // 51.2 MB
  size_t cntBytes  = ((size_t)N_NODES * sizeof(float) + 255) & ~(size_t)255;

  float* summed = (float*)(ws);
  float* cnt    = (float*)(ws + featBytes);
  float* h1     = (float*)(ws + featBytes + cntBytes);
  float* h2     = (float*)(ws + featBytes + cntBytes + featBytes);

  const int gridScatter = (N_EDGES * 32) / 256;   // 200000 blocks
  const int gridGemm    = N_NODES / 16;           // 6250 blocks

  const float* hin = x;
  float* houts[3] = {h1, h2, (float*)d_out};

  for (int i = 0; i < 3; ++i) {
    hipMemsetAsync(summed, 0, featBytes, stream);
    hipMemsetAsync(cnt, 0, (size_t)N_NODES * sizeof(float), stream);
    scatter_sum_kernel<<<gridScatter, 256, 0, stream>>>(hin, src, dst, summed, cnt);
    if (i < 2)
      sage_gemm_ln_kernel<0><<<gridGemm, 256, 0, stream>>>(
          summed, cnt, hin, Wl[i], bl[i], Wr[i], gg[i], bb[i], houts[i]);
    else
      sage_gemm_ln_kernel<1><<<gridGemm, 256, 0, stream>>>(
          summed, cnt, hin, Wl[i], bl[i], Wr[i], gg[i], bb[i], houts[i]);
    hin = houts[i];
  }
}